// SpacialAttention2d_41772851920979
// MI455X (gfx1250) — compile-verified
//
#include <hip/hip_runtime.h>

typedef float v2f __attribute__((ext_vector_type(2)));
typedef float v8f __attribute__((ext_vector_type(8)));

#define NB   16
#define NC   256
#define NH   128
#define NW   128
#define HW   (NH * NW)        // 16384
#define CHW  ((size_t)NC * HW)

// ---------------------------------------------------------------------------
// Phase 1: channel-wise mean + max. Pure HBM streaming: 268 MB in, 2 MB out.
// float2 of pixels per lane -> 131072 threads = 4096 waves (fills the machine
// even at ~320 WGPs), each wave-load moves 256 B contiguous. Non-temporal:
// x (268 MB) exceeds L2 (192 MB) and has zero reuse. Unroll 16 -> 16
// outstanding b64 loads per lane (~16 KB in flight per SIMD at 4 waves/SIMD,
// above the ~9 KB latency-bandwidth product needed for 23.3 TB/s).
// ---------------------------------------------------------------------------
__global__ __launch_bounds__(128) void reduce_mean_max(
    const float* __restrict__ x,    // [NB][NC][NH][NW]
    float* __restrict__ meanp,      // [NB][HW]
    float* __restrict__ maxp)       // [NB][HW]
{
    int tid = blockIdx.x * 128 + threadIdx.x;   // 0 .. 131071
    int b   = tid >> 13;                        // 8192 pixel-pairs per image
    int p2  = tid & 8191;

    const float* base = x + (size_t)b * CHW + (size_t)p2 * 2;

    v2f s = {0.0f, 0.0f};
    v2f m = {-3.402823e38f, -3.402823e38f};

    #pragma unroll 16
    for (int c = 0; c < NC; ++c) {
        v2f v = __builtin_nontemporal_load((const v2f*)(base + (size_t)c * HW));
        s += v;
        m = __builtin_elementwise_max(m, v);
    }

    v2f mean = s * (1.0f / 256.0f);
    *(v2f*)(meanp + (size_t)b * HW + (size_t)p2 * 2) = mean;
    *(v2f*)(maxp  + (size_t)b * HW + (size_t)p2 * 2) = m;
}

// ---------------------------------------------------------------------------
// Phase 2: 7x7, 2-in/1-out conv + ReLU as implicit GEMM on the WMMA pipe.
//   V_WMMA_F32_16X16X4_F32 : C[16x16] += A[16x4] x B[4x16]  (fp32 exact)
// A = weights replicated over M, B = im2col patches, N = 16 output pixels.
// K = 98 taps padded to 100 = 25 chunks of 4; lane l / vgpr v owns
// K = 2*(l>=16)+v, N = l%16. Each wave processes TWO 16-pixel tiles
// (2 accumulators) to amortize tap-record loads.
// Branch-free inner loop: int4 LDS tap table {relOfs, dy-3, dx-3, wBits};
// out-of-range -> select offset 0 (base pixel, always valid), load, select 0.
// ---------------------------------------------------------------------------
__global__ __launch_bounds__(256) void conv7x7_wmma(
    const float* __restrict__ feat,  // [2][NB][NH][NW]  (mean plane, max plane)
    const float* __restrict__ w,     // [1][2][7][7] = 98 floats
    float* __restrict__ out)         // [NB][NH][NW]
{
    __shared__ int4 stab[100];       // {relOfs, dy-3, dx-3, weightBits}
    {
        int t = threadIdx.x;
        if (t < 100) {
            int4 rec;
            if (t < 98) {
                int ci = t / 49;
                int r  = t - ci * 49;
                int dy = r / 7;
                int dx = r - dy * 7;
                rec.x = ci * (NB * HW) + (dy - 3) * NW + (dx - 3);
                rec.y = dy - 3;
                rec.z = dx - 3;
                rec.w = __float_as_int(w[t]);
            } else {
                rec.x = 0; rec.y = -1000; rec.z = -1000; rec.w = 0;
            }
            stab[t] = rec;
        }
    }
    __syncthreads();

    int wave = blockIdx.x * 8 + (threadIdx.x >> 5);   // 8192 waves total
    int lane = threadIdx.x & 31;
    int m    = lane & 15;       // pixel within tile (N index)
    int h    = lane >> 4;       // half-wave -> K subgroup

    int b    = wave >> 9;               // 512 waves per image (128 y * 4 groups)
    int rem  = wave & 511;
    int y    = rem >> 2;                // output row
    int x0   = (rem & 3) << 5;          // 32-pixel group start

    int xp0 = x0 + m;                   // tile-0 pixel x
    int xp1 = xp0 + 16;                 // tile-1 pixel x
    const float* base0 = feat + (size_t)b * HW + (size_t)y * NW + xp0;
    const float* base1 = base0 + 16;

    v8f acc0 = {0.0f, 0.0f, 0.0f, 0.0f, 0.0f, 0.0f, 0.0f, 0.0f};
    v8f acc1 = {0.0f, 0.0f, 0.0f, 0.0f, 0.0f, 0.0f, 0.0f, 0.0f};

    for (int chunk = 0; chunk < 25; ++chunk) {
        int tp = chunk * 4 + h * 2;
        int4 r0 = stab[tp];
        int4 r1 = stab[tp + 1];

        v2f A = { __int_as_float(r0.w), __int_as_float(r1.w) };

        bool vy0 = (unsigned)(y + r0.y) < (unsigned)NH;
        bool vy1 = (unsigned)(y + r1.y) < (unsigned)NH;

        // tile 0
        bool va = vy0 && ((unsigned)(xp0 + r0.z) < (unsigned)NW);
        bool vb = vy1 && ((unsigned)(xp0 + r1.z) < (unsigned)NW);
        float pa = base0[va ? r0.x : 0];
        float pb = base0[vb ? r1.x : 0];
        v2f B0 = { va ? pa : 0.0f, vb ? pb : 0.0f };

        // tile 1
        bool vc = vy0 && ((unsigned)(xp1 + r0.z) < (unsigned)NW);
        bool vd = vy1 && ((unsigned)(xp1 + r1.z) < (unsigned)NW);
        float pc = base1[vc ? r0.x : 0];
        float pd = base1[vd ? r1.x : 0];
        v2f B1 = { vc ? pc : 0.0f, vd ? pd : 0.0f };

        acc0 = __builtin_amdgcn_wmma_f32_16x16x4_f32(
            false, A, false, B0, (short)0, acc0, false, false);
        acc1 = __builtin_amdgcn_wmma_f32_16x16x4_f32(
            false, A, false, B1, (short)0, acc1, false, false);
    }

    if (lane < 16) {
        float* o = out + (size_t)b * HW + (size_t)y * NW + x0 + m;
        o[0]  = fmaxf(acc0[0], 0.0f);
        o[16] = fmaxf(acc1[0], 0.0f);
    }
}

// ---------------------------------------------------------------------------
extern "C" void kernel_launch(void* const* d_in, const int* in_sizes, int n_in,
                              void* d_out, int out_size, void* d_ws, size_t ws_size,
                              hipStream_t stream) {
    const float* x  = (const float*)d_in[0];   // [16,256,128,128]
    const float* cw = (const float*)d_in[1];   // [1,2,7,7]
    float* out = (float*)d_out;                // [16,1,128,128]
    float* ws  = (float*)d_ws;                 // needs 2*16*16384*4 = 2 MB

    float* meanp = ws;
    float* maxp  = ws + (size_t)NB * HW;

    // Phase 1: 16 * 8192 pixel-pairs -> 131072 threads, 1024 blocks of 128
    reduce_mean_max<<<1024, 128, 0, stream>>>(x, meanp, maxp);
    // Phase 2: 8192 waves (2 tiles each), 8 waves per block -> 1024 blocks
    conv7x7_wmma<<<1024, 256, 0, stream>>>(meanp, cw, out);
}